// SparsePropMaxPool_12077448036560
// MI455X (gfx1250) — compile-verified
//
#include <hip/hip_runtime.h>
#include <stdint.h>

#define B_DIM 32
#define H_DIM 512
#define N_CLIPS 64
#define N_PROPS 2080

// d_out layout (floats): [props_h | ori_map_h | ori_map_mask]
#define OUT1_SIZE ((size_t)B_DIM * N_PROPS * H_DIM)             // 34,078,720
#define OUT2_SIZE ((size_t)B_DIM * H_DIM * N_CLIPS * N_CLIPS)   // 67,108,864
#define OUT3_SIZE ((size_t)B_DIM * N_CLIPS * N_CLIPS)           // 131,072

typedef unsigned int u32x4 __attribute__((ext_vector_type(4)));
typedef int          i32x8 __attribute__((ext_vector_type(8)));
typedef int          i32x4 __attribute__((ext_vector_type(4)));

// ---------- CDNA5 async / tensor helpers ----------
__device__ __forceinline__ uint32_t lds_off_of(const void* p) {
  // low 32 bits of a generic pointer into __shared__ == wave-relative LDS byte offset
  return (uint32_t)(uintptr_t)p;
}

__device__ __forceinline__ void async_load_lds_b64(uint32_t lds_off, const void* gaddr) {
#if defined(__gfx1250__)
  asm volatile("global_load_async_to_lds_b64 %0, %1, off"
               :: "v"(lds_off), "v"(gaddr) : "memory");
#else
  (void)lds_off; (void)gaddr;
#endif
}

__device__ __forceinline__ void wait_async0() {
#if defined(__gfx1250__)
#if __has_builtin(__builtin_amdgcn_s_wait_asynccnt)
  __builtin_amdgcn_s_wait_asynccnt(0);
#else
  asm volatile("s_wait_asynccnt 0x0" ::: "memory");
#endif
#endif
}

__device__ __forceinline__ void wait_tensor0() {
#if defined(__gfx1250__)
#if __has_builtin(__builtin_amdgcn_s_wait_tensorcnt)
  __builtin_amdgcn_s_wait_tensorcnt(0);
#else
  asm volatile("s_wait_tensorcnt 0x0" ::: "memory");
#endif
#endif
}

// ---------- validity of a (s, d=e-s) cell in the pooled map ----------
__device__ __forceinline__ bool valid_pos(int s, int d) {
  if (d < 0) return false;
  if (d <= 15) return true;                                            // scale 0
  if (d >= 17 && d <= 31 && (d & 1) && ((s & 1) == 0)) return true;    // scale 1
  if (d >= 35 && (((d - 35) & 3) == 0) && ((s & 3) == 0)) return true; // scale 2
  return false;
}

// ---------- Kernel A: fill ori_map_h (one wave32 per (b,h) row) ----------
__global__ __launch_bounds__(256) void map_fill_kernel(const float* __restrict__ x,
                                                       float* __restrict__ out2) {
  __shared__ __align__(16) float tbl[8][7][64];   // per-wave: level0 = x row, levels 1..6
  const int wave = threadIdx.x >> 5;
  const int lane = threadIdx.x & 31;
  const int row  = blockIdx.x * 8 + wave;         // row = b*512 + h

  const float* src = x + (size_t)row * 64;
  float* t0 = &tbl[wave][0][0];
  async_load_lds_b64(lds_off_of(t0 + lane * 2), src + lane * 2);
  wait_async0();
  __syncthreads();

  for (int p = 1; p <= 6; ++p) {
    const int half = 1 << (p - 1);
    const float* prev = &tbl[wave][p - 1][0];
    float* cur = &tbl[wave][p][0];
#pragma unroll
    for (int k = 0; k < 2; ++k) {
      int s = lane + k * 32;
      int s2 = s + half; if (s2 > 63) s2 = 63;
      cur[s] = fmaxf(prev[s], prev[s2]);
    }
    __syncthreads();
  }

  const float (*T)[64] = tbl[wave];
  float4* dst = (float4*)(out2 + (size_t)row * 4096);
#pragma unroll 4
  for (int it = 0; it < 32; ++it) {
    int idx4 = it * 32 + lane;
    int s  = idx4 >> 4;
    int e0 = (idx4 & 15) * 4;
    float4 v;
    float* vp = &v.x;
#pragma unroll
    for (int j = 0; j < 4; ++j) {
      int e = e0 + j;
      int d = e - s;
      float val = 0.0f;
      if (valid_pos(s, d)) {
        int L = d + 1;
        int p = 31 - __clz(L);
        val = fmaxf(T[p][s], T[p][e - (1 << p) + 1]);
      }
      vp[j] = val;
    }
    dst[idx4] = v;
  }
}

// ---------- Kernel B: props_h[b,p,h] via TDM-staged LDS sparse tables ----------
// block = (b, 64 h-rows); answers all 2080 props with O(1) range-max queries.
__global__ __launch_bounds__(256) void props_kernel(const float* __restrict__ x,
                                                    const long long* __restrict__ props,
                                                    float* __restrict__ out1) {
  __shared__ __align__(16) float T0[64 * 64];        // level 0: compact 64 rows x 64 (TDM target)
  __shared__ __align__(16) float TL[6][64 * 65];     // levels 1..6, stride 65 (conflict-free)
  __shared__ int2 pr[N_PROPS];

  const int b   = blockIdx.x >> 3;
  const int h0  = (blockIdx.x & 7) * 64;
  const int tid = threadIdx.x;
  const float* xtile = x + ((size_t)b * H_DIM + h0) * 64;   // contiguous 16 KB

  // ---- stage x tile into LDS ----
#if defined(__gfx1250__) && __has_builtin(__builtin_amdgcn_tensor_load_to_lds)
  if (tid < 32) {
    uint64_t ga = (uint64_t)(uintptr_t)xtile;
    u32x4 g0 = { 1u,                                   // count=1 (valid descriptor)
                 lds_off_of(T0),                       // lds_addr
                 (uint32_t)ga,                         // global_addr[31:0]
                 ((uint32_t)(ga >> 32) & 0x01FFFFFFu) | (2u << 30) };  // addr[56:32] | type=2
    i32x8 g1 = { (2 << 16),      // data_size = 4B; no pad/iterate/multicast
                 (4096 << 16),   // tensor_dim0 = 4096 (low 16 bits at [79:64... bits 31:16])
                 0,              // tensor_dim0 hi / tensor_dim1 lo
                 (4096 << 16),   // tile_dim0 = 4096
                 0,              // tile_dim1 = 0 (unused), tile_dim2 = 0
                 4096,           // tensor_dim0_stride = 4096
                 0, 0 };
    i32x4 gz = { 0, 0, 0, 0 };
#if __has_include(<hip/amd_detail/amd_gfx1250_TDM.h>)
    i32x8 gz8 = { 0, 0, 0, 0, 0, 0, 0, 0 };
    __builtin_amdgcn_tensor_load_to_lds(g0, g1, gz, gz, gz8, 0);
#else
    __builtin_amdgcn_tensor_load_to_lds(g0, g1, gz, gz, 0);
#endif
    wait_tensor0();
  }
#else
  for (int i = 0; i < 8; ++i) {
    int g = i * 256 + tid;                    // float2 index, 0..2047
    async_load_lds_b64(lds_off_of(&T0[g * 2]), xtile + (size_t)g * 2);
  }
  wait_async0();
#endif

  // ---- preload props as (s, e-1) pairs ----
  for (int i = tid; i < N_PROPS; i += 256) {
    int s  = (int)props[2 * i];
    int e1 = (int)props[2 * i + 1] - 1;
    pr[i] = make_int2(s, e1);
  }
  __syncthreads();

  // ---- build doubling-max levels 1..6 ----
  for (int i = 0; i < 16; ++i) {             // level 1 from compact T0
    int cell = i * 256 + tid;
    int hl = cell >> 6, s = cell & 63;
    int s2 = s + 1; if (s2 > 63) s2 = 63;
    TL[0][hl * 65 + s] = fmaxf(T0[hl * 64 + s], T0[hl * 64 + s2]);
  }
  __syncthreads();
  for (int p = 2; p <= 6; ++p) {
    const int half = 1 << (p - 1);
    const float* prev = &TL[p - 2][0];
    float* cur = &TL[p - 1][0];
    for (int i = 0; i < 16; ++i) {
      int cell = i * 256 + tid;
      int hl = cell >> 6, s = cell & 63;
      int s2 = s + half; if (s2 > 63) s2 = 63;
      cur[hl * 65 + s] = fmaxf(prev[hl * 65 + s], prev[hl * 65 + s2]);
    }
    __syncthreads();
  }

  // ---- answer all props: thread (pgrp, hl) ----
  const int hl   = tid & 63;
  const int pgrp = tid >> 6;                  // 0..3
  for (int it = 0; it < N_PROPS / 4; ++it) {  // 520 iterations
    int p = it * 4 + pgrp;
    int2 se = pr[p];
    int s = se.x, e1 = se.y, d = e1 - s;
    float v = 0.0f;
    if (valid_pos(s, d)) {
      if (d == 0) {
        v = T0[hl * 64 + s];
      } else {
        int lp = 31 - __clz(d + 1);           // >= 1
        const float* lv = &TL[lp - 1][0];
        v = fmaxf(lv[hl * 65 + s], lv[hl * 65 + (e1 - (1 << lp) + 1)]);
      }
    }
    out1[((size_t)b * N_PROPS + p) * H_DIM + h0 + hl] = v;
  }
}

// ---------- Kernel C: base mask pattern ----------
__global__ void mask_base_kernel(float* __restrict__ out3) {
  int idx = blockIdx.x * blockDim.x + threadIdx.x;   // 0 .. 131071
  int rem = idx & 4095;
  int s = rem >> 6;
  int e = rem & 63;
  out3[idx] = valid_pos(s, e - s) ? 1.0f : 0.0f;
}

// ---------- Kernel D: scatter props into mask (all b) ----------
__global__ void mask_props_kernel(const long long* __restrict__ props,
                                  float* __restrict__ out3) {
  int t = blockIdx.x * blockDim.x + threadIdx.x;     // 0 .. 66559
  if (t >= N_PROPS * 32) return;
  int p = t >> 5;
  int b = t & 31;
  int s  = (int)props[2 * p];
  int e1 = (int)props[2 * p + 1] - 1;
  out3[(size_t)b * 4096 + s * 64 + e1] = 1.0f;
}

extern "C" void kernel_launch(void* const* d_in, const int* in_sizes, int n_in,
                              void* d_out, int out_size, void* d_ws, size_t ws_size,
                              hipStream_t stream) {
  (void)in_sizes; (void)n_in; (void)out_size; (void)d_ws; (void)ws_size;
  const float*     x     = (const float*)d_in[0];
  const long long* props = (const long long*)d_in[1];
  float* out  = (float*)d_out;
  float* out1 = out;
  float* out2 = out + OUT1_SIZE;
  float* out3 = out + OUT1_SIZE + OUT2_SIZE;

  map_fill_kernel<<<(B_DIM * H_DIM) / 8, 256, 0, stream>>>(x, out2);
  props_kernel<<<B_DIM * 8, 256, 0, stream>>>(x, props, out1);
  mask_base_kernel<<<(int)(OUT3_SIZE / 256), 256, 0, stream>>>(out3);
  mask_props_kernel<<<(N_PROPS * 32) / 256, 256, 0, stream>>>(props, out3);
}